// YOLOv3Loss_54924041781950
// MI455X (gfx1250) — compile-verified
//
#include <hip/hip_runtime.h>
#include <math.h>

#define NUM_CLASSES   80
#define IMG_SIZE      416.0f
#define IGNORE_THRESH 0.5f
#define LAMBDA_COORD  5.0f
#define LAMBDA_NOOBJ  0.5f
#define MAX_T         1024   // max B*N targets cached in LDS (here 32*20=640)

typedef __attribute__((ext_vector_type(2))) float v2f;
typedef __attribute__((ext_vector_type(8))) float v8f;

__device__ __forceinline__ float softplusf(float x) {
    // numerically stable log(1+exp(x))
    return fmaxf(x, 0.0f) + log1pf(expf(-fabsf(x)));
}
__device__ __forceinline__ float bcef(float x, float t) {
    return softplusf(x) - x * t;
}

// Full wave32 sum via V_WMMA_F32_16X16X4_F32:
//   A (16x4): lane L in [0,15]  -> A[L][0]    = p_L   (VGPR0), A[L][1]=0 (VGPR1)
//             lane L in [16,31] -> A[L-16][2] = p_L,  A[L-16][3]=0
//   B (4x16) = ones  =>  D[m][n] = p_m + p_{m+16}  (broadcast over n)
//   C/D layout: VGPR v holds M=v (lanes 0-15) / M=v+8 (lanes 16-31),
//   so sum of own 8 acc VGPRs gives half-sums; one xor-16 shuffle completes it.
__device__ __forceinline__ float wave_reduce_wmma(float p) {
    v2f a;  a[0] = p;    a[1] = 0.0f;
    v2f bo; bo[0] = 1.0f; bo[1] = 1.0f;
    v8f c = {};
    v8f d = __builtin_amdgcn_wmma_f32_16x16x4_f32(
        /*neg_a=*/false, a, /*neg_b=*/false, bo,
        /*c_mod=*/(short)0, c, /*reuse_a=*/false, /*reuse_b=*/false);
    float s = d[0] + d[1] + d[2] + d[3] + d[4] + d[5] + d[6] + d[7];
    s += __shfl_xor(s, 16, 32);
    return s;
}

__global__ void yolo_init_acc(float* acc) {
    if (threadIdx.x < 2) acc[threadIdx.x] = 0.0f;
}

__global__ __launch_bounds__(256) void yolo_scale_kernel(
        const float* __restrict__ pred,      // [B,255,H,W]
        const float* __restrict__ targets,   // [B,N,5]
        const unsigned char* __restrict__ valid, // [B,N] bool
        const float* __restrict__ anchors,   // [9,2] pixels
        float* __restrict__ acc,             // acc[0]=loss sum, acc[1]=num_pos
        int B, int N, int H, int W, int abase) {
    __shared__ int           ls_ji[MAX_T];   // (gj<<16)|gi, or -1 if invalid
    __shared__ float4        ls_box[MAX_T];  // tx, ty, tw, th
    __shared__ int           ls_cls[MAX_T];
    __shared__ unsigned char ls_best[MAX_T];
    __shared__ unsigned char ls_ign[MAX_T];  // bit k: iou(anchor k) > 0.5

    // normalized anchors for this scale (broadcast, L0-cached)
    float aw[3], ah[3];
#pragma unroll
    for (int k = 0; k < 3; ++k) {
        aw[k] = anchors[(abase + k) * 2 + 0] * (1.0f / IMG_SIZE);
        ah[k] = anchors[(abase + k) * 2 + 1] * (1.0f / IMG_SIZE);
    }

    // ---- Phase 1: per-target metadata into LDS ----
    const int NT = B * N;
    for (int g = threadIdx.x; g < NT; g += blockDim.x) {
        int ji = -1, cls = 0;
        unsigned char best = 0, ign = 0;
        float4 box = make_float4(0.f, 0.f, 0.f, 0.f);
        if (valid[g]) {
            const float* t = targets + (size_t)g * 5;
            float gx = t[0], gy = t[1], gw = t[2], gh = t[3];
            cls = (int)t[4];
            int gi = min((int)(gx * (float)W), W - 1);
            int gj = min((int)(gy * (float)H), H - 1);
            ji = (gj << 16) | gi;
            float biou = -1.0f;
#pragma unroll
            for (int k = 0; k < 3; ++k) {
                float inter = fminf(gw, aw[k]) * fminf(gh, ah[k]);
                float uni   = gw * gh + aw[k] * ah[k] - inter + 1e-16f;
                float iou   = inter / uni;
                if (iou > biou) { biou = iou; best = (unsigned char)k; }
                if (iou > IGNORE_THRESH) ign |= (unsigned char)(1u << k);
            }
            box.x = gx * (float)W - (float)gi;
            box.y = gy * (float)H - (float)gj;
            box.z = logf(gw / aw[best]);
            box.w = logf(gh / ah[best]);
        }
        ls_ji[g] = ji; ls_box[g] = box; ls_cls[g] = cls;
        ls_best[g] = best; ls_ign[g] = ign;
    }
    __syncthreads();

    // ---- Phase 2: one thread per objectness cell (b, a, j, i) ----
    const int HW = H * W;
    const int total = B * 3 * HW;
    const int idx = blockIdx.x * blockDim.x + threadIdx.x;

    float partial = 0.0f;
    if (idx < total) {
        const int i = idx % W;
        const int j = (idx / W) % H;
        const int a = (idx / HW) % 3;
        const int b = idx / (HW * 3);

        // only 3 of 255 channel planes are ever streamed (85x traffic cut)
        const float* pc = pred + ((size_t)b * 255 + (size_t)a * 85) * HW
                               + (size_t)j * W + i;
        const float conf = pc[(size_t)4 * HW];

        const int key = (j << 16) | i;
        int pos_n = -1;
        bool ignore = false;
        const int tb = b * N;
        for (int n = 0; n < N; ++n) {
            const int t = tb + n;
            if (ls_ji[t] == key) {
                if (ls_ign[t] & (1u << a)) ignore = true;
                if (ls_best[t] == (unsigned char)a) pos_n = t; // last write wins
            }
        }

        if (pos_n >= 0) {
            const float4 box = ls_box[pos_n];
            const float px = pc[0];
            const float py = pc[(size_t)1 * HW];
            const float pw = pc[(size_t)2 * HW];
            const float ph = pc[(size_t)3 * HW];
            float l = bcef(px, box.x) + bcef(py, box.y);
            const float dw = pw - box.z, dh = ph - box.w;
            l += dw * dw + dh * dh;
            float loss = LAMBDA_COORD * l + softplusf(-conf);
            const int cls = ls_cls[pos_n];
            const float* pcl = pc + (size_t)5 * HW;
            float lc = 0.0f;
            for (int c = 0; c < NUM_CLASSES; ++c) {
                const float x = pcl[(size_t)c * HW];
                lc += softplusf(x) - ((c == cls) ? x : 0.0f);
            }
            partial = loss + lc;
            atomicAdd(&acc[1], 1.0f);          // num_pos (rare)
        } else if (!ignore) {
            partial = LAMBDA_NOOBJ * softplusf(conf);
        }
    }

    // all threads reach here (EXEC all-1s) -> WMMA wave reduction
    const float wsum = wave_reduce_wmma(partial);
    if ((threadIdx.x & 31) == 0) atomicAdd(&acc[0], wsum);
}

__global__ void yolo_finalize(const float* acc, float* out, float Bf) {
    const float t = acc[0], p = acc[1];
    out[0] = (p > 0.0f) ? (t / p) : (t / Bf);
}

extern "C" void kernel_launch(void* const* d_in, const int* in_sizes, int n_in,
                              void* d_out, int out_size, void* d_ws, size_t ws_size,
                              hipStream_t stream) {
    const float* preds[3] = { (const float*)d_in[0],
                              (const float*)d_in[1],
                              (const float*)d_in[2] };
    const float* targets = (const float*)d_in[3];
    const unsigned char* valid = (const unsigned char*)d_in[4]; // jnp bool = 1 byte
    const float* anchors = (const float*)d_in[5];

    const int dims[3] = { 13, 26, 52 };
    const int B = in_sizes[0] / (255 * 13 * 13);   // 32
    const int N = in_sizes[3] / (B * 5);           // 20

    float* acc = (float*)d_ws;                     // 2 floats of scratch
    yolo_init_acc<<<1, 32, 0, stream>>>(acc);

    for (int s = 0; s < 3; ++s) {
        const int H = dims[s], W = dims[s];
        const int abase = 6 - 3 * s;               // ANCHOR_MASKS[s][0]
        const int total = B * 3 * H * W;
        const int grid = (total + 255) / 256;
        yolo_scale_kernel<<<grid, 256, 0, stream>>>(
            preds[s], targets, valid, anchors, acc, B, N, H, W, abase);
    }

    yolo_finalize<<<1, 1, 0, stream>>>(acc, (float*)d_out, (float)B);
}